// BidirectionalLSTM_69698729280012
// MI455X (gfx1250) — compile-verified
//
#include <hip/hip_runtime.h>
#include <hip/hip_bf16.h>

#define S_LEN 2048
#define B_TOT 4096
#define H_SZ  8

typedef __attribute__((ext_vector_type(2))) float v2f;
typedef __attribute__((ext_vector_type(8))) float v8f;

// Branchless fast activations: keep the serial recurrence path short.
// v_rcp_f32 / v_exp_f32 / v_tanh_f32 are single trans ops on CDNA5.
__device__ __forceinline__ float frcp(float v) {
    return __builtin_amdgcn_rcpf(v);
}

__device__ __forceinline__ float fsigmoid(float v) {
    return frcp(1.0f + __expf(-v));
}

__device__ __forceinline__ float ftanh(float v) {
#if __has_builtin(__builtin_amdgcn_tanhf)
    return __builtin_amdgcn_tanhf(v);          // native V_TANH_F32
#else
    return 1.0f - 2.0f * frcp(1.0f + __expf(2.0f * v));
#endif
}

// CDNA5 split dependency counter wait: order LDS traffic within the wave
// (no workgroup barrier needed; 1 wave per block owns its LDS slice).
__device__ __forceinline__ void lds_fence() {
    asm volatile("s_wait_dscnt 0x0" ::: "memory");
}

__device__ __forceinline__ v8f wmma4(v2f a, v2f b, v8f c) {
    // D(16x16 f32) = A(16x4 f32) * B(4x16 f32) + C
    return __builtin_amdgcn_wmma_f32_16x16x4_f32(
        /*neg_a=*/false, a, /*neg_b=*/false, b,
        /*c_mod=*/(short)0, c, /*reuse_a=*/false, /*reuse_b=*/false);
}

__global__ __launch_bounds__(32)
void bilstm_fused(const float* __restrict__ x,
                  const float* __restrict__ w_ih_f, const float* __restrict__ w_hh_f,
                  const float* __restrict__ b_ih_f, const float* __restrict__ b_hh_f,
                  const float* __restrict__ w_ih_b, const float* __restrict__ w_hh_b,
                  const float* __restrict__ b_ih_b, const float* __restrict__ b_hh_b,
                  const float* __restrict__ w_out, const float* __restrict__ b_out,
                  const int* __restrict__ future_p,
                  float* __restrict__ out)
{
    // gates transpose buffer: [dir][batch-row m][gate n], row stride 40 (pad, 16B-aligned rows)
    __shared__ __align__(16) float gbuf[2][16][40];
    // h relayout buffer: [dir][batch-row m][hidden j]
    __shared__ __align__(16) float hbuf[2][16][8];

    const int lane = threadIdx.x & 31;
    const int half = lane >> 4;        // lane half selects K/M sub-blocks per WMMA layout
    const int nn   = lane & 15;        // A: M-row / B: N-col / C: N-col / elementwise: batch row
    const int bb   = blockIdx.x * 16;  // batch tile base
    const int fut  = future_p[0];
    const int j0   = 4 * half;         // this lane's hidden quad for elementwise work

    const float* wih[2] = { w_ih_f, w_ih_b };
    const float* whh[2] = { w_hh_f, w_hh_b };
    const float* bih[2] = { b_ih_f, b_ih_b };
    const float* bhh[2] = { b_hh_f, b_hh_b };

    // Combined weight matrix Wc[K=12][N=32] per direction, held in B-matrix layout:
    // lane col n = nn + 16*tile;  K = 4*chunk + 2*half + v
    //   K<8 : w_hh[n][K]   (h @ w_hh.T)
    //   K==8: w_ih[n]      (x projection, I==1)
    //   K==9: b_ih[n]+b_hh[n]
    //   else: 0
    v2f wmat[2][2][3];
    #pragma unroll
    for (int d = 0; d < 2; ++d)
        #pragma unroll
        for (int t = 0; t < 2; ++t)
            #pragma unroll
            for (int c = 0; c < 3; ++c) {
                const int n = nn + 16 * t;
                float comp[2];
                #pragma unroll
                for (int v = 0; v < 2; ++v) {
                    const int K = 4 * c + 2 * half + v;
                    float val = 0.0f;
                    if (K < 8)       val = whh[d][n * H_SZ + K];
                    else if (K == 8) val = wih[d][n];
                    else if (K == 9) val = bih[d][n] + bhh[d][n];
                    comp[v] = val;
                }
                v2f w = { comp[0], comp[1] };
                wmat[d][t][c] = w;
            }

    // output projection weights for this lane's hidden quad (fwd: w_out[0:8], bwd: w_out[8:16])
    float wo[2][4];
    #pragma unroll
    for (int jj = 0; jj < 4; ++jj) {
        wo[0][jj] = w_out[j0 + jj];
        wo[1][jj] = w_out[8 + j0 + jj];
    }
    const float bo = b_out[0];

    // recurrent state: h held in A-matrix layout (chunk0: j=2*half+v, chunk1: j=4+2*half+v),
    // cell state c held per-lane for cells (m=nn, j=j0..j0+3)
    v2f a0[2], a1[2];
    a0[0] = a0[1] = (v2f){0.0f, 0.0f};
    a1[0] = a1[1] = (v2f){0.0f, 0.0f};
    float cst[2][4] = {{0,0,0,0},{0,0,0,0}};

    // x prefetch for step 0 (fwd reads x[i], bwd reads x[(future - i) mod S])
    float xv[2];
    xv[0] = x[(size_t)(0 & (S_LEN - 1)) * B_TOT + bb + nn];
    xv[1] = x[(size_t)(fut & (S_LEN - 1)) * B_TOT + bb + nn];

    const float kone = half ? 0.0f : 1.0f;   // loop-invariant "1" slot of A chunk2

    for (int ii = 0; ii < S_LEN; ii += 4) {
        float ov[4];
        #pragma unroll
        for (int u = 0; u < 4; ++u) {
            const int i = ii + u;

            // prefetch next step's x (overlaps WMMA chain)
            float xnxt[2];
            xnxt[0] = x[(size_t)((i + 1) & (S_LEN - 1)) * B_TOT + bb + nn];
            xnxt[1] = x[(size_t)((fut - i - 1) & (S_LEN - 1)) * B_TOT + bb + nn];

            // 4 independent chains of 3 accumulating WMMAs: gates = [h, x, 1, 0, 0] * Wc
            v8f acc0[2], acc1[2];
            #pragma unroll
            for (int d = 0; d < 2; ++d) {
                const v2f a2 = { xv[d] * kone, kone };
                v8f z = {0,0,0,0,0,0,0,0};
                v8f t0 = wmma4(a0[d], wmat[d][0][0], z);
                t0 = wmma4(a1[d], wmat[d][0][1], t0);
                t0 = wmma4(a2,    wmat[d][0][2], t0);   // gates i,f (n=0..15)
                acc0[d] = t0;
                v8f t1 = wmma4(a0[d], wmat[d][1][0], z);
                t1 = wmma4(a1[d], wmat[d][1][1], t1);
                t1 = wmma4(a2,    wmat[d][1][2], t1);   // gates g,o (n=16..31)
                acc1[d] = t1;
            }

            // C-layout -> [m][n] transpose through LDS (D row m = v + 8*half, col n = nn)
            #pragma unroll
            for (int d = 0; d < 2; ++d)
                #pragma unroll
                for (int v = 0; v < 8; ++v) {
                    gbuf[d][v + 8 * half][nn]      = acc0[d][v];
                    gbuf[d][v + 8 * half][16 + nn] = acc1[d][v];
                }
            lds_fence();

            // elementwise cell update: lane owns (m=nn, j=j0..j0+3); torch gate order i,f,g,o
            float partial = 0.0f;
            #pragma unroll
            for (int d = 0; d < 2; ++d) {
                float hq[4];
                #pragma unroll
                for (int jj = 0; jj < 4; ++jj) {
                    const int j = j0 + jj;
                    const float ig = fsigmoid(gbuf[d][nn][j]);
                    const float fg = fsigmoid(gbuf[d][nn][8 + j]);
                    const float gg = ftanh(gbuf[d][nn][16 + j]);
                    const float og = fsigmoid(gbuf[d][nn][24 + j]);
                    const float cc = fg * cst[d][jj] + ig * gg;
                    cst[d][jj] = cc;
                    const float hh = og * ftanh(cc);
                    hq[jj] = hh;
                    partial += hh * wo[d][jj];
                }
                *(float4*)&hbuf[d][nn][j0] = make_float4(hq[0], hq[1], hq[2], hq[3]);
            }
            lds_fence();

            // h back into A-matrix layout for the next step
            #pragma unroll
            for (int d = 0; d < 2; ++d) {
                a0[d] = *(const v2f*)&hbuf[d][nn][2 * half];
                a1[d] = *(const v2f*)&hbuf[d][nn][4 + 2 * half];
            }

            // out[b, i] = w_out . cat(hf, hb) + b_out ; combine the two j-halves of each m
            const float s = partial + __shfl_xor(partial, 16, 32);
            ov[u] = s + bo;

            xv[0] = xnxt[0];
            xv[1] = xnxt[1];
        }
        if (lane < 16) {
            *(float4*)&out[(size_t)(bb + nn) * S_LEN + ii] =
                make_float4(ov[0], ov[1], ov[2], ov[3]);
        }
    }
}

extern "C" void kernel_launch(void* const* d_in, const int* in_sizes, int n_in,
                              void* d_out, int out_size, void* d_ws, size_t ws_size,
                              hipStream_t stream) {
    (void)in_sizes; (void)n_in; (void)out_size; (void)d_ws; (void)ws_size;
    const float* xp     = (const float*)d_in[0];
    const float* w_ih_f = (const float*)d_in[1];
    const float* w_hh_f = (const float*)d_in[2];
    const float* b_ih_f = (const float*)d_in[3];
    const float* b_hh_f = (const float*)d_in[4];
    const float* w_ih_b = (const float*)d_in[5];
    const float* w_hh_b = (const float*)d_in[6];
    const float* b_ih_b = (const float*)d_in[7];
    const float* b_hh_b = (const float*)d_in[8];
    const float* w_out  = (const float*)d_in[9];
    const float* b_out  = (const float*)d_in[10];
    const int*   future = (const int*)d_in[11];
    float* out = (float*)d_out;

    dim3 grid(B_TOT / 16);   // 256 waves: one 16-batch tile, both directions per wave
    dim3 block(32);          // wave32
    bilstm_fused<<<grid, block, 0, stream>>>(
        xp, w_ih_f, w_hh_f, b_ih_f, b_hh_f,
        w_ih_b, w_hh_b, b_ih_b, b_hh_b,
        w_out, b_out, future, out);
}